// MPNN_26774826123550
// MI455X (gfx1250) — compile-verified
//
#include <hip/hip_runtime.h>
#include <hip/hip_bf16.h>
#include <stdint.h>

typedef __attribute__((ext_vector_type(2))) float v2f;
typedef __attribute__((ext_vector_type(8))) float v8f;

#define D 64

// ---------------------------------------------------------------------------
// Fold + pack weights into WMMA B-fragment layout.
//   W1 = Wm @ Wu_top (64x64), W2 = Wu_bot (64x64), b1 = bm @ Wu_top (64)
// Packed layout (per matrix, 4096 floats):
//   entry t = ((colTile*16 + kGroup)*32 + lane), value = {B[k0][n], B[k0+1][n]}
//   where khi = (lane>=16)?2:0, k0 = 4*kGroup + khi, n = 16*colTile + (lane&15)
// This makes every B-fragment load in the hot loop a single aligned b64.
// Work items: [0,2048) -> W1p, [2048,4096) -> W2p, [4096,4160) -> b1.
// ---------------------------------------------------------------------------
__global__ void fold_pack_weights(const float* __restrict__ Wm,
                                  const float* __restrict__ bm,
                                  const float* __restrict__ Wu,
                                  float* __restrict__ W1p,
                                  float* __restrict__ W2p,
                                  float* __restrict__ b1) {
    int tid = blockIdx.x * blockDim.x + threadIdx.x;
    if (tid < 4096) {
        int t    = tid & 2047;
        int lane = t & 31;
        int kg   = (t >> 5) & 15;
        int c    = t >> 9;                       // column tile 0..3
        int khi  = (lane >= 16) ? 2 : 0;
        int k0   = 4 * kg + khi;
        int n    = 16 * c + (lane & 15);
        if (tid < 2048) {
            // W1p: fold Wm @ Wu_top directly into packed slots
            float s0 = 0.f, s1 = 0.f;
            #pragma unroll 8
            for (int j = 0; j < D; ++j) {
                float w = Wu[j * D + n];
                s0 += Wm[k0 * D + j] * w;
                s1 += Wm[(k0 + 1) * D + j] * w;
            }
            W1p[t * 2 + 0] = s0;
            W1p[t * 2 + 1] = s1;
        } else {
            // W2p: permutation copy of Wu bottom half
            W2p[t * 2 + 0] = Wu[(D + k0) * D + n];
            W2p[t * 2 + 1] = Wu[(D + k0 + 1) * D + n];
        }
    } else if (tid < 4160) {
        int c2 = tid - 4096;
        float s = 0.f;
        #pragma unroll 8
        for (int k = 0; k < D; ++k) s += bm[k] * Wu[k * D + c2];
        b1[c2] = s;
    }
}

// ---------------------------------------------------------------------------
// Zero h (N*64) and deg (N) -- stored contiguously.
// ---------------------------------------------------------------------------
__global__ void zero_f32(float* __restrict__ p, long n) {
    long i = (long)blockIdx.x * blockDim.x + threadIdx.x;
    if (i < n) p[i] = 0.f;
}

// ---------------------------------------------------------------------------
// Scatter: h[dst] += x[src] for E edges plus N self loops; deg[dst] += 1.
// 16 threads per (virtual) edge, each moves a float4 of the 64-float row.
// x and h are L2-resident (12.8 MB each on a 192 MB L2), so the 51M f32
// atomics per layer stay on-chip.
// ---------------------------------------------------------------------------
__global__ void scatter_add(const float* __restrict__ x,
                            const long long* __restrict__ src,
                            const long long* __restrict__ dst,
                            float* __restrict__ h,
                            float* __restrict__ deg,
                            int E, int N) {
    long idx = (long)blockIdx.x * blockDim.x + threadIdx.x;
    long e = idx >> 4;                 // virtual edge id
    int  g = (int)(idx & 15) * 4;      // feature group (float4)
    long total = (long)E + N;
    if (e >= total) return;
    long s, d;
    if (e < E) { s = (long)src[e]; d = (long)dst[e]; }
    else       { s = d = e - E; }      // self loop
    const float4 v = *(const float4*)(x + s * D + g);
    float* hp = h + d * D + g;
    atomicAdd(hp + 0, v.x);
    atomicAdd(hp + 1, v.y);
    atomicAdd(hp + 2, v.z);
    atomicAdd(hp + 3, v.w);
    if (g == 0) atomicAdd(deg + d, 1.0f);
}

// ---------------------------------------------------------------------------
// Fused node update via f32 WMMA:
//   out = relu( h @ W1 + x @ W2 + deg (x) b1 + bu )
// One block = 128 threads = 4 waves; block owns a 16-node row tile, each wave
// owns one 16-column tile (4 x 16 = 64 output cols).
// V_WMMA_F32_16X16X4_F32 layouts (ISA 7.12.2):
//   A 16x4:  lanes 0-15 row M=lane, {K0,K1}; lanes 16-31 same rows, {K2,K3}
//   B 4x16:  lanes 0-15 col N=lane, {K0,K1}; lanes 16-31 same cols, {K2,K3}
//   C/D:     VGPR v -> row v (lanes 0-15) / row v+8 (lanes 16-31), col = lane&15
// B fragments come from the packed layout -> single b64 per step.
// ---------------------------------------------------------------------------
__global__ void node_update_wmma(const float* __restrict__ h,
                                 const float* __restrict__ x,
                                 const float* __restrict__ W1p,
                                 const float* __restrict__ W2p,
                                 const float* __restrict__ b1,
                                 const float* __restrict__ deg,
                                 const float* __restrict__ bu,
                                 float* __restrict__ out,
                                 int N) {
    const int wave = threadIdx.x >> 5;       // 0..3 -> column tile
    const int lane = threadIdx.x & 31;
    const int m0 = blockIdx.x * 16;
    if (m0 >= N) return;                     // wave-uniform: EXEC stays all-ones
    const int n0 = wave * 16;
    const int lm = lane & 15;
    const int khi = (lane >= 16) ? 2 : 0;    // K sub-offset for this half-wave

    // Per-wave packed B pointers: fragment kg lives at [kg*32 + lane]
    const v2f* B1 = (const v2f*)W1p + (wave * 16) * 32 + lane;
    const v2f* B2 = (const v2f*)W2p + (wave * 16) * 32 + lane;

    v8f acc = {};

    const float* Arow = h + (long)(m0 + lm) * D;
    #pragma unroll
    for (int kg = 0; kg < 16; ++kg) {
        v2f a;
        a.x = Arow[4 * kg + khi + 0];
        a.y = Arow[4 * kg + khi + 1];
        v2f b = B1[kg * 32];
        acc = __builtin_amdgcn_wmma_f32_16x16x4_f32(
            /*neg_a=*/false, a, /*neg_b=*/false, b,
            /*c_mod=*/(short)0, acc, /*reuse_a=*/false, /*reuse_b=*/false);
    }

    const float* Xrow = x + (long)(m0 + lm) * D;
    #pragma unroll
    for (int kg = 0; kg < 16; ++kg) {
        v2f a;
        a.x = Xrow[4 * kg + khi + 0];
        a.y = Xrow[4 * kg + khi + 1];
        v2f b = B2[kg * 32];
        acc = __builtin_amdgcn_wmma_f32_16x16x4_f32(
            false, a, false, b, (short)0, acc, false, false);
    }

    // Epilogue: bias + deg-scaled folded message bias + ReLU
    const float bias1 = b1[n0 + lm];
    const float biasu = bu[n0 + lm];
    const int mbase = m0 + ((lane >= 16) ? 8 : 0);
    #pragma unroll
    for (int v = 0; v < 8; ++v) {
        int m = mbase + v;
        float r = acc[v] + deg[m] * bias1 + biasu;
        out[(long)m * D + n0 + lm] = (r > 0.f) ? r : 0.f;
    }
}

// ---------------------------------------------------------------------------
// Driver
// ---------------------------------------------------------------------------
extern "C" void kernel_launch(void* const* d_in, const int* in_sizes, int n_in,
                              void* d_out, int out_size, void* d_ws, size_t ws_size,
                              hipStream_t stream) {
    const float* x0 = (const float*)d_in[0];
    const long long* ei = (const long long*)d_in[1];
    const int N = in_sizes[0] / D;       // 50000
    const int E = in_sizes[1] / 2;       // 800000

    const long long* src = ei;
    const long long* dst = ei + E;

    // Workspace carve-up (floats): bufA | bufB | h | deg | W1p | W2p | b1
    float* ws   = (float*)d_ws;
    float* bufA = ws;
    float* bufB = bufA + (size_t)N * D;
    float* hbuf = bufB + (size_t)N * D;
    float* deg  = hbuf + (size_t)N * D;   // contiguous with h for one-shot zero
    float* W1p  = deg + N;
    float* W2p  = W1p + D * D;
    float* b1   = W2p + D * D;

    const long zn = (long)N * D + N;                       // h + deg elements
    const long sc_threads = ((long)E + N) * 16;            // 16 thr / edge

    for (int l = 0; l < 3; ++l) {
        const float* xc = (l == 0) ? x0 : ((l == 1) ? bufA : bufB);
        float* xo = (l == 0) ? bufA : ((l == 1) ? bufB : (float*)d_out);
        const float* Wm = (const float*)d_in[2 + 4 * l];
        const float* bm = (const float*)d_in[3 + 4 * l];
        const float* Wu = (const float*)d_in[4 + 4 * l];
        const float* bu = (const float*)d_in[5 + 4 * l];

        fold_pack_weights<<<(4160 + 255) / 256, 256, 0, stream>>>(
            Wm, bm, Wu, W1p, W2p, b1);

        zero_f32<<<(int)((zn + 255) / 256), 256, 0, stream>>>(hbuf, zn);

        scatter_add<<<(int)((sc_threads + 255) / 256), 256, 0, stream>>>(
            xc, src, dst, hbuf, deg, E, N);

        node_update_wmma<<<(N + 15) / 16, 128, 0, stream>>>(
            hbuf, xc, W1p, W2p, b1, deg, bu, xo, N);
    }
}